// PlanetoidBunch_18648747999740
// MI455X (gfx1250) — compile-verified
//
#include <hip/hip_runtime.h>

typedef __attribute__((ext_vector_type(2))) float v2f;
typedef __attribute__((ext_vector_type(8))) float v8f;

#define FEAT 64
#define EPB  64        // edges per block
#define T1   128       // threads in GEMM kernel (4 waves)

// ---------------- k0: zero the accumulation output ----------------
__global__ void zero_kernel(float4* __restrict__ out, int n4) {
    int i = blockIdx.x * blockDim.x + threadIdx.x;
    if (i < n4) out[i] = make_float4(0.f, 0.f, 0.f, 0.f);
}

// ---------------- k1: fused edge feature + WMMA f32 GEMM ----------------
// Xl[e, :] = ((edge_list[u]-edge_list[v])^2) @ W^T + b
__global__ __launch_bounds__(T1)
void edge_gemm_kernel(const float* __restrict__ edge_list,  // [N,64]
                      const int*   __restrict__ X1,         // [E,2]
                      const float* __restrict__ W,          // [64,64]
                      const float* __restrict__ bias,       // [64]
                      float*       __restrict__ Xl,         // [E,64] scratch
                      int E) {
    __shared__ float Wl[FEAT][66];   // padded: bank = (2*row + k) % 64, conflict-free
    __shared__ float Xe[EPB][66];
    __shared__ int   uv[EPB][2];

    const int t     = threadIdx.x;
    const int ebase = blockIdx.x * EPB;

    // Stage W into LDS (64x64 f32 = 16 KB), float4 loads
    for (int i = t; i < FEAT * 16; i += T1) {
        int r = i >> 4, c = (i & 15) << 2;
        float4 w4 = *(const float4*)(W + r * FEAT + c);
        Wl[r][c + 0] = w4.x; Wl[r][c + 1] = w4.y;
        Wl[r][c + 2] = w4.z; Wl[r][c + 3] = w4.w;
    }
    // Stage endpoint indices
    if (t < EPB) {
        int e = ebase + t;
        int u = 0, v = 0;
        if (e < E) { u = X1[2 * e]; v = X1[2 * e + 1]; }
        uv[t][0] = u; uv[t][1] = v;
    }
    __syncthreads();

    // Compute Xe = (x_u - x_v)^2 into LDS, float4 granularity
    for (int i = t; i < EPB * 16; i += T1) {
        int el = i >> 4, c = (i & 15) << 2;
        int e  = ebase + el;
        float4 q = make_float4(0.f, 0.f, 0.f, 0.f);
        if (e < E) {
            const float4 a = *(const float4*)(edge_list + (size_t)uv[el][0] * FEAT + c);
            const float4 b = *(const float4*)(edge_list + (size_t)uv[el][1] * FEAT + c);
            float dx = a.x - b.x, dy = a.y - b.y, dz = a.z - b.z, dw = a.w - b.w;
            q = make_float4(dx * dx, dy * dy, dz * dz, dw * dw);
        }
        Xe[el][c + 0] = q.x; Xe[el][c + 1] = q.y;
        Xe[el][c + 2] = q.z; Xe[el][c + 3] = q.w;
    }
    __syncthreads();

    // Each wave owns a 16-edge tile; full 64-col output via 4 N-tiles.
    const int wave  = t >> 5;
    const int lane  = t & 31;
    const int m     = lane & 15;   // A-row / B-col / D-col within tile
    const int khalf = lane >> 4;   // selects K pair {0,1} vs {2,3} of each 4-K block
    const int arow  = wave * 16 + m;

    v8f acc[4] = {{}, {}, {}, {}};

#pragma unroll
    for (int kb = 0; kb < 16; ++kb) {
        const int k = kb * 4 + khalf * 2;
        v2f a = *(const v2f*)&Xe[arow][k];           // A: 16x4 f32 fragment
#pragma unroll
        for (int nt = 0; nt < 4; ++nt) {
            const int jn = nt * 16 + m;              // output feature column
            v2f bf = *(const v2f*)&Wl[jn][k];        // B[k][j] = W[j][k]
            acc[nt] = __builtin_amdgcn_wmma_f32_16x16x4_f32(
                false, a, false, bf, (short)0, acc[nt], false, false);
        }
    }

    // D layout: VGPR r -> M = r + 8*khalf, N = lane&15. Add bias, store.
#pragma unroll
    for (int nt = 0; nt < 4; ++nt) {
        const int j  = nt * 16 + m;
        const float bj = bias[j];
#pragma unroll
        for (int r = 0; r < 8; ++r) {
            const int e = ebase + wave * 16 + r + 8 * khalf;
            if (e < E) Xl[(size_t)e * FEAT + j] = acc[nt][r] + bj;
        }
    }
}

// ---------------- k2: COO scatter-add, 16 lanes per nnz ----------------
__global__ void scatter_kernel(const float* __restrict__ Xl,
                               const int*   __restrict__ rows,
                               const int*   __restrict__ cols,
                               const float* __restrict__ vals,
                               float*       __restrict__ out,
                               int nnz) {
    long long gid = (long long)blockIdx.x * blockDim.x + threadIdx.x;
    long long i   = gid >> 4;
    if (i >= nnz) return;
    const int c4 = (int)(gid & 15) << 2;
    const int r  = rows[i];
    const int c  = cols[i];
    const float v = vals[i];
    const float4 x = *(const float4*)(Xl + (size_t)c * FEAT + c4);
    float* o = out + (size_t)r * FEAT + c4;
    unsafeAtomicAdd(o + 0, v * x.x);
    unsafeAtomicAdd(o + 1, v * x.y);
    unsafeAtomicAdd(o + 2, v * x.z);
    unsafeAtomicAdd(o + 3, v * x.w);
}

// ---------------- k3: PReLU in place ----------------
__global__ void prelu_kernel(float* __restrict__ out,
                             const float* __restrict__ slope_p, int n) {
    int i = blockIdx.x * blockDim.x + threadIdx.x;
    if (i >= n) return;
    const float s = slope_p[0];
    const float x = out[i];
    out[i] = (x >= 0.f) ? x : s * x;
}

extern "C" void kernel_launch(void* const* d_in, const int* in_sizes, int n_in,
                              void* d_out, int out_size, void* d_ws, size_t ws_size,
                              hipStream_t stream) {
    const float* edge_list = (const float*)d_in[0];   // [N,64]
    const int*   X1        = (const int*)  d_in[1];   // [E,2]
    const float* W         = (const float*)d_in[2];   // [64,64]
    const float* bias      = (const float*)d_in[3];   // [64]
    const float* prelu_w   = (const float*)d_in[4];   // [1]
    const int*   b1_rows   = (const int*)  d_in[5];   // [NNZ]
    const int*   b1_cols   = (const int*)  d_in[6];   // [NNZ]
    const float* b1_vals   = (const float*)d_in[7];   // [NNZ]

    const int E   = in_sizes[1] / 2;
    const int NNZ = in_sizes[5];
    const int nOut = out_size;            // N * 64 floats

    float* Xl = (float*)d_ws;             // [E,64] scratch

    // k0: zero output accumulator
    {
        int n4 = nOut / 4;
        zero_kernel<<<(n4 + 255) / 256, 256, 0, stream>>>((float4*)d_out, n4);
    }
    // k1: fused edge features + WMMA GEMM
    {
        int blocks = (E + EPB - 1) / EPB;
        edge_gemm_kernel<<<blocks, T1, 0, stream>>>(edge_list, X1, W, bias, Xl, E);
    }
    // k2: scatter-add
    {
        long long total = (long long)NNZ * 16;
        int blocks = (int)((total + 255) / 256);
        scatter_kernel<<<blocks, 256, 0, stream>>>(Xl, b1_rows, b1_cols, b1_vals,
                                                   (float*)d_out, NNZ);
    }
    // k3: PReLU
    {
        prelu_kernel<<<(nOut + 255) / 256, 256, 0, stream>>>((float*)d_out, prelu_w, nOut);
    }
}